// DCTSCE_59906203845017
// MI455X (gfx1250) — compile-verified
//
#include <hip/hip_runtime.h>

// MI455X / gfx1250 implementation.
// Phase 1: f32 WMMA (V_WMMA_F32_16X16X4_F32) 2D-DCT of 8x8 blocks, 2x2 blocks per
//          16x16 wave tile, coefficient workspace + per-(n,c,j) min/max via
//          monotone-key uint atomics.
// Phase 2: 403MB streaming reconstruction with nontemporal b128 stores
//          (pure HBM-store-bound: ~17us roofline at 23.3 TB/s).

typedef float v2f __attribute__((ext_vector_type(2)));
typedef float v4f __attribute__((ext_vector_type(4)));
typedef float v8f __attribute__((ext_vector_type(8)));

#define NIMG 24          // N*C = 8*3
#define IMGW 256
#define COEF_FLOATS (NIMG * 64 * 1024)   // [img][j][by*32+bx]
#define TILES_PER_WAVE 4                 // 64 waves/img * 4 tiles = 256 tiles

__constant__ float c_D[64] = {
  0.35355339059327373f, 0.35355339059327373f, 0.35355339059327373f, 0.35355339059327373f,
  0.35355339059327373f, 0.35355339059327373f, 0.35355339059327373f, 0.35355339059327373f,
  0.4903926402016152f,  0.4157348061512726f,  0.2777851165098011f,  0.09754516100806412f,
 -0.09754516100806412f,-0.2777851165098011f, -0.4157348061512726f, -0.4903926402016152f,
  0.46193976625564337f, 0.19134171618254492f,-0.19134171618254492f,-0.46193976625564337f,
 -0.46193976625564337f,-0.19134171618254492f, 0.19134171618254492f, 0.46193976625564337f,
  0.4157348061512726f, -0.09754516100806412f,-0.4903926402016152f, -0.2777851165098011f,
  0.2777851165098011f,  0.4903926402016152f,  0.09754516100806412f,-0.4157348061512726f,
  0.35355339059327373f,-0.35355339059327373f,-0.35355339059327373f, 0.35355339059327373f,
  0.35355339059327373f,-0.35355339059327373f,-0.35355339059327373f, 0.35355339059327373f,
  0.2777851165098011f, -0.4903926402016152f,  0.09754516100806412f, 0.4157348061512726f,
 -0.4157348061512726f, -0.09754516100806412f, 0.4903926402016152f, -0.2777851165098011f,
  0.19134171618254492f,-0.46193976625564337f, 0.46193976625564337f,-0.19134171618254492f,
 -0.19134171618254492f, 0.46193976625564337f,-0.46193976625564337f, 0.19134171618254492f,
  0.09754516100806412f,-0.2777851165098011f,  0.4157348061512726f, -0.4903926402016152f,
  0.4903926402016152f, -0.4157348061512726f,  0.2777851165098011f, -0.09754516100806412f
};

// inverse zig-zag: j = c_INVZZ[k*8 + l]
__constant__ unsigned char c_INVZZ[64] = {
   0, 2, 3, 9,10,20,21,35,
   1, 4, 8,11,19,22,34,36,
   5, 7,12,18,23,33,37,48,
   6,13,17,24,32,38,47,49,
  14,16,25,31,39,46,50,57,
  15,26,30,40,45,51,56,58,
  27,29,41,44,52,55,59,62,
  28,42,43,53,54,60,61,63
};
__constant__ unsigned char c_K1[64] = {
  0,1,0,0,1,2,3,2,1,0,0,1,2,3,4,5,4,3,2,1,0,0,1,2,3,4,5,6,7,6,5,4,
  3,2,1,0,1,2,3,4,5,6,7,7,6,5,4,3,2,3,4,5,6,7,7,6,5,4,5,6,7,7,6,7};
__constant__ unsigned char c_K2[64] = {
  0,0,1,2,1,0,0,1,2,3,4,3,2,1,0,0,1,2,3,4,5,6,5,4,3,2,1,0,0,1,2,3,
  4,5,6,7,7,6,5,4,3,2,1,2,3,4,5,6,7,7,6,5,4,3,4,5,6,7,7,6,5,6,7,7};

__device__ __forceinline__ unsigned fkey(float f) {
  unsigned u = __float_as_uint(f);
  return (u & 0x80000000u) ? ~u : (u ^ 0x80000000u);
}
__device__ __forceinline__ float funkey(unsigned k) {
  unsigned u = (k & 0x80000000u) ? (k ^ 0x80000000u) : ~k;
  return __uint_as_float(u);
}

// ---------------- Phase 1: WMMA DCT + coef store + min/max ----------------
__global__ void __launch_bounds__(128)
dct_coef_kernel(const float* __restrict__ x, float* __restrict__ coefs,
                unsigned* __restrict__ mnKey, unsigned* __restrict__ mxKey) {
  const int img  = blockIdx.x;                 // n*3 + c
  const int wave = threadIdx.x >> 5;
  const int lane = threadIdx.x & 31;
  const int wg   = blockIdx.y * 4 + wave;      // 0..63
  const int h    = lane >> 4;                  // half (0: lanes 0-15, 1: 16-31)
  const int lr   = lane & 15;

  // Block-diagonal Dd = diag(D,D) fragments.
  // A-layout of Dd chunk k  == B-layout of Dd^T chunk k (both = Dd[lr, 4k+2h+v]).
  v2f dfrag[4];
#pragma unroll
  for (int k = 0; k < 4; ++k) {
#pragma unroll
    for (int v = 0; v < 2; ++v) {
      int cc = 4 * k + 2 * h + v;
      float val = 0.0f;
      if (lr < 8 && cc < 8) val = c_D[lr * 8 + cc];
      if (lr >= 8 && cc >= 8) val = c_D[(lr - 8) * 8 + (cc - 8)];
      dfrag[k][v] = val;
    }
  }

  float mnAcc[8], mxAcc[8];
#pragma unroll
  for (int r = 0; r < 8; ++r) { mnAcc[r] = __builtin_inff(); mxAcc[r] = -__builtin_inff(); }

  const float* imgp = x + (size_t)img * (IMGW * IMGW);
  float* cimg = coefs + (size_t)img * 65536;

  for (int i = 0; i < TILES_PER_WAVE; ++i) {
    int t  = wg * TILES_PER_WAVE + i;          // 0..255 tiles (16x16 tiles of 16x16 px)
    int ty = t >> 4, tx = t & 15;

    // A fragments of X: aX[k][v] = X[lr, 4k+2h+v]  (lane reads one image row)
    const v4f* rp = (const v4f*)(imgp + (size_t)(ty * 16 + lr) * IMGW + tx * 16);
    v2f aX[4];
#pragma unroll
    for (int k = 0; k < 4; ++k) {
      v4f f = rp[k];
      v2f lo; lo[0] = f.x; lo[1] = f.y;
      v2f hi; hi[0] = f.z; hi[1] = f.w;
      aX[k] = h ? hi : lo;
    }

    // T1 = X * Dd^T   (4 chained K=4 f32 WMMAs)
    v8f t1 = {0.f,0.f,0.f,0.f,0.f,0.f,0.f,0.f};
#pragma unroll
    for (int k = 0; k < 4; ++k)
      t1 = __builtin_amdgcn_wmma_f32_16x16x4_f32(false, aX[k], false, dfrag[k],
                                                 (short)0, t1, false, false);

    // C-layout (t1[r] = T1[8h+r, lr]) -> B-layout chunks via lane^16 exchange
    float oth[8];
#pragma unroll
    for (int r = 0; r < 8; ++r) oth[r] = __shfl_xor(t1[r], 16, 32);

    v2f bT[4];
#pragma unroll
    for (int k = 0; k < 4; ++k) {
#pragma unroll
      for (int v = 0; v < 2; ++v) {
        int reg = 4 * (k & 1) + 2 * h + v;          // (4k+2h+v) & 7
        bool useOther = h ? (k < 2) : (k >= 2);     // row's owning half != h
        bT[k][v] = useOther ? oth[reg] : t1[reg];
      }
    }

    // Y = Dd * T1 : per-quadrant 8x8 DCT coefficients
    v8f y = {0.f,0.f,0.f,0.f,0.f,0.f,0.f,0.f};
#pragma unroll
    for (int k = 0; k < 4; ++k)
      y = __builtin_amdgcn_wmma_f32_16x16x4_f32(false, dfrag[k], false, bT[k],
                                                (short)0, y, false, false);

    // y[r] = coef(k=r, l=lr&7) of block (by=2ty+h, bx=2tx+(lr>>3))
    int by = 2 * ty + h, bx = 2 * tx + (lr >> 3), l = lr & 7;
#pragma unroll
    for (int r = 0; r < 8; ++r) {
      int j = c_INVZZ[r * 8 + l];
      float v = y[r];
      cimg[j * 1024 + by * 32 + bx] = v;
      mnAcc[r] = fminf(mnAcc[r], v);
      mxAcc[r] = fmaxf(mxAcc[r], v);
    }
  }

  // lanes {l, l^8, l^16, l^24} share the same j for a given r
#pragma unroll
  for (int r = 0; r < 8; ++r) {
    mnAcc[r] = fminf(mnAcc[r], __shfl_xor(mnAcc[r], 8, 32));
    mnAcc[r] = fminf(mnAcc[r], __shfl_xor(mnAcc[r], 16, 32));
    mxAcc[r] = fmaxf(mxAcc[r], __shfl_xor(mxAcc[r], 8, 32));
    mxAcc[r] = fmaxf(mxAcc[r], __shfl_xor(mxAcc[r], 16, 32));
  }
  if (lane < 8) {
#pragma unroll
    for (int r = 0; r < 8; ++r) {
      int j = c_INVZZ[r * 8 + lane];
      atomicMin(&mnKey[img * 64 + j], fkey(mnAcc[r]));
      atomicMax(&mxKey[img * 64 + j], fkey(mxAcc[r]));
    }
  }
}

// ---------------- Phase 2: streaming reconstruction (store-bound) ----------------
__global__ void __launch_bounds__(256)
recon_kernel(const float* __restrict__ coefs, const unsigned* __restrict__ mnKey,
             const unsigned* __restrict__ mxKey, float* __restrict__ out) {
  int bid  = blockIdx.x;              // ((n*64 + j)*3 + c)*16 + rg
  int rg   = bid & 15;
  int rest = bid >> 4;
  int c    = rest % 3; rest /= 3;
  int j    = rest & 63;
  int n    = rest >> 6;
  int img  = n * 3 + c;

  float cmn = funkey(mnKey[img * 64 + j]);
  float cmx = funkey(mxKey[img * 64 + j]);
  int k1 = c_K1[j], k2 = c_K2[j];

  // basis extremes (exact: fl() is monotone, so product-set extremes = extreme products)
  float rmn = 1e30f, rmx = -1e30f, qmn = 1e30f, qmx = -1e30f;
#pragma unroll
  for (int i = 0; i < 8; ++i) {
    float dv = c_D[k1 * 8 + i]; rmn = fminf(rmn, dv); rmx = fmaxf(rmx, dv);
    float ev = c_D[k2 * 8 + i]; qmn = fminf(qmn, ev); qmx = fmaxf(qmx, ev);
  }
  float b00 = rmn * qmn, b01 = rmn * qmx, b10 = rmx * qmn, b11 = rmx * qmx;
  float bmn = fminf(fminf(b00, b01), fminf(b10, b11));
  float bmx = fmaxf(fmaxf(b00, b01), fmaxf(b10, b11));
  float p00 = cmn * bmn, p01 = cmn * bmx, p10 = cmx * bmn, p11 = cmx * bmx;
  float mn = fminf(fminf(p00, p01), fminf(p10, p11));
  float mx = fmaxf(fmaxf(p00, p01), fmaxf(p10, p11));

  float a  = 510.0f / (mx - mn + 1e-5f);
  float bb = -(a * mn + 255.0f);

  int tid     = threadIdx.x;
  int rl      = tid >> 4;             // 0..15 -> row within group
  int colBase = (tid & 15) * 16;      // 16 consecutive columns (2 blocks)
  int yrow    = rg * 16 + rl;
  int by      = yrow >> 3, p = yrow & 7;
  int bx0     = colBase >> 3;

  float rowv = c_D[k1 * 8 + p];
  const float* cb = coefs + (size_t)img * 65536 + j * 1024 + by * 32;
  float s0 = a * cb[bx0] * rowv;
  float s1 = a * cb[bx0 + 1] * rowv;

  float cv[8];
#pragma unroll
  for (int q = 0; q < 8; ++q) cv[q] = c_D[k2 * 8 + q];

  float* op = out + ((size_t)(n * 192 + j * 3 + c) * IMGW + yrow) * IMGW + colBase;
  v4f o0 = { fabsf(fmaf(s0, cv[0], bb)), fabsf(fmaf(s0, cv[1], bb)),
             fabsf(fmaf(s0, cv[2], bb)), fabsf(fmaf(s0, cv[3], bb)) };
  v4f o1 = { fabsf(fmaf(s0, cv[4], bb)), fabsf(fmaf(s0, cv[5], bb)),
             fabsf(fmaf(s0, cv[6], bb)), fabsf(fmaf(s0, cv[7], bb)) };
  v4f o2 = { fabsf(fmaf(s1, cv[0], bb)), fabsf(fmaf(s1, cv[1], bb)),
             fabsf(fmaf(s1, cv[2], bb)), fabsf(fmaf(s1, cv[3], bb)) };
  v4f o3 = { fabsf(fmaf(s1, cv[4], bb)), fabsf(fmaf(s1, cv[5], bb)),
             fabsf(fmaf(s1, cv[6], bb)), fabsf(fmaf(s1, cv[7], bb)) };
  __builtin_nontemporal_store(o0, (v4f*)op);
  __builtin_nontemporal_store(o1, (v4f*)(op + 4));
  __builtin_nontemporal_store(o2, (v4f*)(op + 8));
  __builtin_nontemporal_store(o3, (v4f*)(op + 12));
}

extern "C" void kernel_launch(void* const* d_in, const int* in_sizes, int n_in,
                              void* d_out, int out_size, void* d_ws, size_t ws_size,
                              hipStream_t stream) {
  const float* x = (const float*)d_in[0];
  float* out = (float*)d_out;

  float* coefs     = (float*)d_ws;                                  // 6.29 MB
  unsigned* mnKey  = (unsigned*)((char*)d_ws + (size_t)COEF_FLOATS * 4);
  unsigned* mxKey  = mnKey + NIMG * 64;

  (void)hipMemsetAsync(mnKey, 0xFF, NIMG * 64 * sizeof(unsigned), stream); // key=UINT_MAX
  (void)hipMemsetAsync(mxKey, 0x00, NIMG * 64 * sizeof(unsigned), stream); // key=0

  dct_coef_kernel<<<dim3(NIMG, 16), 128, 0, stream>>>(x, coefs, mnKey, mxKey);
  recon_kernel<<<dim3(8 * 64 * 3 * 16), 256, 0, stream>>>(coefs, mnKey, mxKey, out);
}